// SSIMLoss_58059367907771
// MI455X (gfx1250) — compile-verified
//
#include <hip/hip_runtime.h>

typedef __attribute__((ext_vector_type(2))) float v2f;
typedef __attribute__((ext_vector_type(8))) float v8f;

#define WMMA4(a, b, c) \
  __builtin_amdgcn_wmma_f32_16x16x4_f32(false, (a), false, (b), (short)0, (c), false, false)

constexpr int IMG_H = 512;
constexpr int IMG_W = 512;
constexpr int NCHAN = 3;
constexpr int NBATCH = 32;
constexpr float SSIM_C1 = 0.01f * 0.01f;
constexpr float SSIM_C2 = 0.03f * 0.03f;
constexpr float SSIM_EPS = 1e-12f;

// ws layout: [0..31] per-sample ssim accumulators, [32..42] separable 1-D gaussian
__global__ void ssim_prep(const float* __restrict__ w2d, float* __restrict__ ws) {
  int t = threadIdx.x;
  if (t < 32) ws[t] = 0.0f;
  if (t >= 32 && t < 43) {
    int i = t - 32;
    float s = 0.0f;
    for (int j = 0; j < 11; ++j) s += w2d[i * 11 + j];  // row sum of outer(g,g) == g[i]
    ws[32 + i] = s;
  }
}

__global__ __launch_bounds__(32)
void ssim_tile(const float* __restrict__ pred, const float* __restrict__ target,
               float* __restrict__ ws) {
  // 26x28 staged input tiles (28 = K padded to multiple of 4);
  // 5 per-plane 32x16 horizontal-conv intermediates
  __shared__ float sx[26 * 28];
  __shared__ float sy[26 * 28];
  __shared__ float sh[5][32 * 16];
  __shared__ float gs[16];

  const int lane = threadIdx.x;
  const int half = lane >> 4;   // which 16-lane half of the wave
  const int m    = lane & 15;   // M index for A/D, N index for B/D
  const int img  = blockIdx.z;  // n*3 + c
  const int row0 = blockIdx.y * 16;
  const int col0 = blockIdx.x * 16;
  const size_t base = (size_t)img * (IMG_H * IMG_W);

  if (lane < 16) gs[lane] = (lane < 11) ? ws[32 + lane] : 0.0f;

  // Stage clamped inputs with 5-halo; out-of-image -> 0 (matches zero padding).
  for (int i = lane; i < 26 * 28; i += 32) {
    int r = i / 28, c = i - r * 28;
    int gr = row0 - 5 + r;
    int gc = col0 - 5 + c;
    float vx = 0.0f, vy = 0.0f;
    if (gr >= 0 && gr < IMG_H && gc >= 0 && gc < IMG_W) {
      float px = pred[base + (size_t)gr * IMG_W + gc];
      float py = target[base + (size_t)gr * IMG_W + gc];
      vx = fminf(fmaxf(px, 0.0f), 1.0f);
      vy = fminf(fmaxf(py, 0.0f), 1.0f);
    }
    sx[i] = vx;
    sy[i] = vy;
  }
  __syncthreads();

  // Banded Gaussian chunks. For horizontal B: Wh[k][n] = g[k-n]; for vertical A:
  // Wv[m][k] = g[k-m]. With n == m == lane&15 and identical K mapping, they coincide.
  v2f W[7];
#pragma unroll
  for (int q = 0; q < 7; ++q) {
#pragma unroll
    for (int r = 0; r < 2; ++r) {
      int kk = 4 * q + r + 2 * half;
      int d = kk - m;
      W[q][r] = (d >= 0 && d <= 10) ? gs[d] : 0.0f;
    }
  }

  // ---- horizontal pass: load each x/y fragment once, drive 5 WMMA chains ----
#pragma unroll
  for (int b = 0; b < 2; ++b) {
    const int row  = b * 16 + m;            // tile row this lane's A element sits in
    const int rowc = row < 26 ? row : 25;   // keep LDS index in-bounds
    const bool live = row < 26;
    v8f hacc[5] = {{}, {}, {}, {}, {}};
#pragma unroll
    for (int q = 0; q < 7; ++q) {
      v2f a[5];
#pragma unroll
      for (int r = 0; r < 2; ++r) {
        int kk = 4 * q + r + 2 * half;      // padded column (K index)
        float px = live ? sx[rowc * 28 + kk] : 0.0f;
        float py = live ? sy[rowc * 28 + kk] : 0.0f;
        a[0][r] = px;
        a[1][r] = py;
        a[2][r] = px * px;
        a[3][r] = py * py;
        a[4][r] = px * py;
      }
      // 5 independent accumulation chains: hides WMMA RAW latency
#pragma unroll
      for (int p = 0; p < 5; ++p) hacc[p] = WMMA4(a[p], W[q], hacc[p]);
    }
    // D layout -> LDS rows b*16 + r + 8*half
#pragma unroll
    for (int p = 0; p < 5; ++p)
#pragma unroll
      for (int r = 0; r < 8; ++r) sh[p][(b * 16 + r + 8 * half) * 16 + m] = hacc[p][r];
  }
  __syncthreads();

  // ---- vertical pass: Out = Wv(16x28) x H(28x16), planes interleaved per chunk ----
  v8f res[5] = {{}, {}, {}, {}, {}};
#pragma unroll
  for (int q = 0; q < 7; ++q) {
#pragma unroll
    for (int p = 0; p < 5; ++p) {
      v2f bb;
#pragma unroll
      for (int r = 0; r < 2; ++r) {
        int kk = 4 * q + r + 2 * half;
        bb[r] = sh[p][kk * 16 + m];
      }
      res[p] = WMMA4(W[q], bb, res[p]);
    }
  }

  // ---- pointwise SSIM + wave32 reduction ----
  float s = 0.0f;
#pragma unroll
  for (int r = 0; r < 8; ++r) {
    float mux = res[0][r], muy = res[1][r];
    float ex2 = res[2][r], ey2 = res[3][r], exy = res[4][r];
    float mux2 = mux * mux, muy2 = muy * muy, muxy = mux * muy;
    float sx2 = ex2 - mux2;
    float sy2 = ey2 - muy2;
    float sxy = exy - muxy;
    float num = (2.0f * muxy + SSIM_C1) * (2.0f * sxy + SSIM_C2);
    float den = (mux2 + muy2 + SSIM_C1) * (sx2 + sy2 + SSIM_C2) + SSIM_EPS;
    s += num / den;
  }
#pragma unroll
  for (int off = 16; off > 0; off >>= 1) s += __shfl_down(s, off, 32);
  if (lane == 0) {
    int nIdx = img / NCHAN;
    atomicAdd(&ws[nIdx], s);
  }
}

__global__ __launch_bounds__(32)
void ssim_final(const float* __restrict__ ws, float* __restrict__ out) {
  int lane = threadIdx.x;
  float v = 1.0f - ws[lane] * (1.0f / (float)(NCHAN * IMG_H * IMG_W));
#pragma unroll
  for (int off = 16; off > 0; off >>= 1) v += __shfl_down(v, off, 32);
  if (lane == 0) out[0] = v * (1.0f / (float)NBATCH);
}

extern "C" void kernel_launch(void* const* d_in, const int* in_sizes, int n_in,
                              void* d_out, int out_size, void* d_ws, size_t ws_size,
                              hipStream_t stream) {
  const float* pred   = (const float*)d_in[0];
  const float* target = (const float*)d_in[1];
  const float* window = (const float*)d_in[2];
  float* ws  = (float*)d_ws;
  float* out = (float*)d_out;

  ssim_prep<<<1, 64, 0, stream>>>(window, ws);

  dim3 grid(IMG_W / 16, IMG_H / 16, NBATCH * NCHAN);
  ssim_tile<<<grid, 32, 0, stream>>>(pred, target, ws);

  ssim_final<<<1, 32, 0, stream>>>(ws, out);
}